// PathfindingGNN_58789512348244
// MI455X (gfx1250) — compile-verified
//
#include <hip/hip_runtime.h>
#include <math.h>

#define FDIM 128
#define NEG_SLOPE 0.2f

typedef __attribute__((ext_vector_type(2))) float v2f;
typedef __attribute__((ext_vector_type(8))) float v8f;

// ---- ordered-float encoding so we can do float max via atomicMax(u32) ----
__device__ __forceinline__ unsigned f2ord(float f) {
  unsigned u = __float_as_uint(f);
  return (u & 0x80000000u) ? ~u : (u | 0x80000000u);
}
__device__ __forceinline__ float ord2f(unsigned u) {
  return (u & 0x80000000u) ? __uint_as_float(u & 0x7fffffffu) : __uint_as_float(~u);
}

// ---------------------------------------------------------------------------
// C[N x 128] = A[N x 128] * B[128 x 128], fp32 WMMA 16x16x4.
// One wave computes a full 16-row x 128-col stripe: 8 independent 16x16
// accumulators. Per K-step the wave loads ONE A fragment (reused by all 8
// column tiles) and 8 B fragments, then issues 8 *independent* WMMAs
// back-to-back to keep the matrix pipe full (no D->C serial chain between
// them). N must be a multiple of 16 (100000 = 6250*16).
//
// A-frag layout (ISA 7.12.2, 32-bit A 16x4): lane m = lane&15,
//   VGPR0 = K+0 (lanes 0-15) / K+2 (lanes 16-31), VGPR1 = K+1 / K+3
//   -> per lane: two consecutive K elements starting at K + 2*(lane>>4).
// B-frag (4x16, rows striped across lanes): lane n = lane&15, same K split.
// C/D: VGPR i holds row (i + 8*(lane>>4)), col lane&15.
// ---------------------------------------------------------------------------
__global__ void gemm_wmma_f32(const float* __restrict__ A,
                              const float* __restrict__ B,
                              float* __restrict__ C, int N) {
  const int wpb = blockDim.x >> 5;
  int tm = blockIdx.x * wpb + (threadIdx.x >> 5);  // 16-row stripe index
  if (tm * 16 >= N) return;                        // wave-uniform

  int lane  = threadIdx.x & 31;
  int half  = lane >> 4;        // 0 or 1
  int khalf = half << 1;        // 0 or 2
  int l16   = lane & 15;

  const float* arow  = A + (size_t)(tm * 16 + l16) * FDIM;  // row m, walk K
  const float* bbase = B + l16;                             // col n within tile

  v8f acc0 = {}, acc1 = {}, acc2 = {}, acc3 = {};
  v8f acc4 = {}, acc5 = {}, acc6 = {}, acc7 = {};

#pragma unroll
  for (int k = 0; k < FDIM; k += 4) {
    v2f a;
    a.x = arow[k + khalf];
    a.y = arow[k + khalf + 1];
    const float* b0 = bbase + (size_t)(k + khalf) * FDIM;
    const float* b1 = bbase + (size_t)(k + khalf + 1) * FDIM;
    v2f b;
#define GAT_WMMA_STEP(T, ACC)                                                  \
    b.x = b0[(T) * 16];                                                        \
    b.y = b1[(T) * 16];                                                        \
    ACC = __builtin_amdgcn_wmma_f32_16x16x4_f32(false, a, false, b,            \
                                                (short)0, ACC, false, false);
    GAT_WMMA_STEP(0, acc0)
    GAT_WMMA_STEP(1, acc1)
    GAT_WMMA_STEP(2, acc2)
    GAT_WMMA_STEP(3, acc3)
    GAT_WMMA_STEP(4, acc4)
    GAT_WMMA_STEP(5, acc5)
    GAT_WMMA_STEP(6, acc6)
    GAT_WMMA_STEP(7, acc7)
#undef GAT_WMMA_STEP
  }

  int cm = tm * 16 + half * 8;
  float* cbase = C + (size_t)cm * FDIM + l16;
#pragma unroll
  for (int i = 0; i < 8; ++i) {
    float* cr = cbase + (size_t)i * FDIM;
    cr[0 * 16] = acc0[i];
    cr[1 * 16] = acc1[i];
    cr[2 * 16] = acc2[i];
    cr[3 * 16] = acc3[i];
    cr[4 * 16] = acc4[i];
    cr[5 * 16] = acc5[i];
    cr[6 * 16] = acc6[i];
    cr[7 * 16] = acc7[i];
  }
}

// ---------------------------------------------------------------------------
// Per-node attention scores: a_src[i] = h[i,:]·att_s, a_dst[i] = h[i,:]·att_d.
// One wave per node; each lane handles 4 contiguous features.
// ---------------------------------------------------------------------------
__global__ void attn_scores(const float* __restrict__ h,
                            const float* __restrict__ att_s,
                            const float* __restrict__ att_d,
                            float* __restrict__ as_out,
                            float* __restrict__ ad_out, int N) {
  int wave = blockIdx.x * (blockDim.x >> 5) + (threadIdx.x >> 5);
  if (wave >= N) return;
  int lane = threadIdx.x & 31;
  float4 hv = ((const float4*)(h + (size_t)wave * FDIM))[lane];
  float4 sv = ((const float4*)att_s)[lane];
  float4 dv = ((const float4*)att_d)[lane];
  float ps = hv.x * sv.x + hv.y * sv.y + hv.z * sv.z + hv.w * sv.w;
  float pd = hv.x * dv.x + hv.y * dv.y + hv.z * dv.z + hv.w * dv.w;
#pragma unroll
  for (int off = 16; off; off >>= 1) {
    ps += __shfl_xor(ps, off, 32);
    pd += __shfl_xor(pd, off, 32);
  }
  if (lane == 0) { as_out[wave] = ps; ad_out[wave] = pd; }
}

__global__ void init_ms(unsigned* __restrict__ m, float* __restrict__ s, int N) {
  int i = blockIdx.x * blockDim.x + threadIdx.x;
  if (i >= N) return;
  m[i] = f2ord(-INFINITY);
  s[i] = 0.0f;
}

__global__ void zero_f4(float4* __restrict__ p, int n4) {
  int i = blockIdx.x * blockDim.x + threadIdx.x;
  if (i >= n4) return;
  p[i] = make_float4(0.f, 0.f, 0.f, 0.f);
}

// edges [0,E) from edge_index, edges [E, E+N) are self-loops i->i
__global__ void edge_logits(const int* __restrict__ src, const int* __restrict__ dst,
                            const float* __restrict__ as, const float* __restrict__ ad,
                            float* __restrict__ e, unsigned* __restrict__ m,
                            int E, int N) {
  int i = blockIdx.x * blockDim.x + threadIdx.x;
  if (i >= E + N) return;
  int s, d;
  if (i < E) { s = src[i]; d = dst[i]; } else { s = d = i - E; }
  float v = as[s] + ad[d];
  v = (v > 0.f) ? v : NEG_SLOPE * v;  // leaky relu
  e[i] = v;
  atomicMax(m + d, f2ord(v));
}

__global__ void edge_exp(const int* __restrict__ dst, float* __restrict__ e,
                         const unsigned* __restrict__ m, float* __restrict__ ssum,
                         int E, int N) {
  int i = blockIdx.x * blockDim.x + threadIdx.x;
  if (i >= E + N) return;
  int d = (i < E) ? dst[i] : (i - E);
  float ex = __expf(e[i] - ord2f(m[d]));
  e[i] = ex;
  atomicAdd(ssum + d, ex);
}

// One wave per edge: agg[d,:] += (e[i]/ssum[d]) * h[s,:]
__global__ void edge_agg(const int* __restrict__ src, const int* __restrict__ dst,
                         const float* __restrict__ e, const float* __restrict__ ssum,
                         const float* __restrict__ h, float* __restrict__ agg,
                         int E, int N) {
  int i = blockIdx.x * (blockDim.x >> 5) + (threadIdx.x >> 5);
  if (i >= E + N) return;
  int s, d;
  if (i < E) { s = src[i]; d = dst[i]; } else { s = d = i - E; }
  float alpha = e[i] / ssum[d];
  int lane = threadIdx.x & 31;
  float4 hv = ((const float4*)(h + (size_t)s * FDIM))[lane];
  float* ap = agg + (size_t)d * FDIM + lane * 4;
  atomicAdd(ap + 0, alpha * hv.x);
  atomicAdd(ap + 1, alpha * hv.y);
  atomicAdd(ap + 2, alpha * hv.z);
  atomicAdd(ap + 3, alpha * hv.w);
}

// agg[i,:] += bias; optional relu. One thread per float4.
__global__ void bias_act(float* __restrict__ agg, const float* __restrict__ b,
                         int N, int do_relu) {
  int i = blockIdx.x * blockDim.x + threadIdx.x;
  int n4 = N * (FDIM / 4);
  if (i >= n4) return;
  int f4 = i & (FDIM / 4 - 1);
  float4 v = ((float4*)agg)[i];
  float4 bv = ((const float4*)b)[f4];
  v.x += bv.x; v.y += bv.y; v.z += bv.z; v.w += bv.w;
  if (do_relu) {
    v.x = fmaxf(v.x, 0.f); v.y = fmaxf(v.y, 0.f);
    v.z = fmaxf(v.z, 0.f); v.w = fmaxf(v.w, 0.f);
  }
  ((float4*)agg)[i] = v;
}

// One wave per ORIGINAL edge: out[i] = [h[s];h[d]] · Wp + bp
__global__ void edge_pred(const int* __restrict__ src, const int* __restrict__ dst,
                          const float* __restrict__ h, const float* __restrict__ Wp,
                          const float* __restrict__ bp, float* __restrict__ out, int E) {
  int i = blockIdx.x * (blockDim.x >> 5) + (threadIdx.x >> 5);
  if (i >= E) return;
  int s = src[i], d = dst[i];
  int lane = threadIdx.x & 31;
  float4 hs = ((const float4*)(h + (size_t)s * FDIM))[lane];
  float4 hd = ((const float4*)(h + (size_t)d * FDIM))[lane];
  float4 ws = ((const float4*)Wp)[lane];
  float4 wd = ((const float4*)(Wp + FDIM))[lane];
  float p = hs.x * ws.x + hs.y * ws.y + hs.z * ws.z + hs.w * ws.w +
            hd.x * wd.x + hd.y * wd.y + hd.z * wd.z + hd.w * wd.w;
#pragma unroll
  for (int off = 16; off; off >>= 1) p += __shfl_xor(p, off, 32);
  if (lane == 0) out[i] = p + bp[0];
}

extern "C" void kernel_launch(void* const* d_in, const int* in_sizes, int n_in,
                              void* d_out, int out_size, void* d_ws, size_t ws_size,
                              hipStream_t stream) {
  const float* x   = (const float*)d_in[0];
  const int*   ei  = (const int*)d_in[1];
  const float* W1  = (const float*)d_in[2];
  const float* as1 = (const float*)d_in[3];
  const float* ad1 = (const float*)d_in[4];
  const float* b1  = (const float*)d_in[5];
  const float* W2  = (const float*)d_in[6];
  const float* as2 = (const float*)d_in[7];
  const float* ad2 = (const float*)d_in[8];
  const float* b2  = (const float*)d_in[9];
  const float* Wp  = (const float*)d_in[10];
  const float* bp  = (const float*)d_in[11];
  float* out = (float*)d_out;

  const int N  = in_sizes[0] / FDIM;     // 100000
  const int E  = in_sizes[1] / 2;        // 1600000
  const int ET = E + N;                  // with self-loops
  const int* src = ei;
  const int* dst = ei + E;

  // workspace partition (floats): buf0/buf1 = N*128 each, then scalars
  float*    buf0 = (float*)d_ws;                 // gemm output h (gathered by edges)
  float*    buf1 = buf0 + (size_t)N * FDIM;      // aggregation target / next input
  float*    asrc = buf1 + (size_t)N * FDIM;
  float*    adst = asrc + N;
  unsigned* mbuf = (unsigned*)(adst + N);
  float*    sbuf = (float*)(mbuf + N);
  float*    ebuf = sbuf + N;                     // E+N edge scratch

  const int TPB = 256;
  const int WPB = TPB / 32;
  const int gemm_waves = N / 16;                 // one wave per 16x128 stripe
  dim3 gemm_grid((gemm_waves + WPB - 1) / WPB);
  dim3 node_wave_grid((N + WPB - 1) / WPB);
  dim3 node_grid((N + TPB - 1) / TPB);
  dim3 edge_grid((ET + TPB - 1) / TPB);
  dim3 edge_wave_grid((ET + WPB - 1) / WPB);
  dim3 feat4_grid((N * (FDIM / 4) + TPB - 1) / TPB);
  dim3 pred_grid((E + WPB - 1) / WPB);

  // ---------------- layer 1: h1 = relu(GAT(x, W1)) ----------------
  gemm_wmma_f32<<<gemm_grid, TPB, 0, stream>>>(x, W1, buf0, N);
  attn_scores<<<node_wave_grid, TPB, 0, stream>>>(buf0, as1, ad1, asrc, adst, N);
  init_ms<<<node_grid, TPB, 0, stream>>>(mbuf, sbuf, N);
  zero_f4<<<feat4_grid, TPB, 0, stream>>>((float4*)buf1, N * (FDIM / 4));
  edge_logits<<<edge_grid, TPB, 0, stream>>>(src, dst, asrc, adst, ebuf, mbuf, E, N);
  edge_exp<<<edge_grid, TPB, 0, stream>>>(dst, ebuf, mbuf, sbuf, E, N);
  edge_agg<<<edge_wave_grid, TPB, 0, stream>>>(src, dst, ebuf, sbuf, buf0, buf1, E, N);
  bias_act<<<feat4_grid, TPB, 0, stream>>>(buf1, b1, N, 1);

  // ---------------- layer 2: h2 = GAT(h1, W2) ----------------
  gemm_wmma_f32<<<gemm_grid, TPB, 0, stream>>>(buf1, W2, buf0, N);
  attn_scores<<<node_wave_grid, TPB, 0, stream>>>(buf0, as2, ad2, asrc, adst, N);
  init_ms<<<node_grid, TPB, 0, stream>>>(mbuf, sbuf, N);
  edge_logits<<<edge_grid, TPB, 0, stream>>>(src, dst, asrc, adst, ebuf, mbuf, E, N);
  edge_exp<<<edge_grid, TPB, 0, stream>>>(dst, ebuf, mbuf, sbuf, E, N);
  // buf1 already consumed by gemm (stream order) -> reuse as layer-2 aggregation
  zero_f4<<<feat4_grid, TPB, 0, stream>>>((float4*)buf1, N * (FDIM / 4));
  edge_agg<<<edge_wave_grid, TPB, 0, stream>>>(src, dst, ebuf, sbuf, buf0, buf1, E, N);
  bias_act<<<feat4_grid, TPB, 0, stream>>>(buf1, b2, N, 0);

  // ---------------- edge predictor on original edges ----------------
  edge_pred<<<pred_grid, TPB, 0, stream>>>(src, dst, buf1, Wp, bp, out, E);
}